// Model_74603581932004
// MI455X (gfx1250) — compile-verified
//
#include <hip/hip_runtime.h>
#include <hip/hip_bf16.h>

#define N_NODES   100000
#define N_EDGES   1200000
#define NODE_DIM  64
#define NUM_LAYERS 3
#define NUM_GRAPHS 256
#define R_HID     128
#define R_OUT     32

typedef __attribute__((ext_vector_type(2))) float v2f;
typedef __attribute__((ext_vector_type(8))) float v8f;

static __device__ __forceinline__ v8f wmma_f32_k4(v2f a, v2f b, v8f c) {
    // D = A(16x4 f32) * B(4x16 f32) + C(16x16 f32)
    return __builtin_amdgcn_wmma_f32_16x16x4_f32(
        /*neg_a=*/false, a, /*neg_b=*/false, b,
        /*c_mod=*/(short)0, c, /*reuse_a=*/false, /*reuse_b=*/false);
}

// ---------------------------------------------------------------------------
// Edge scatter-add: agg[dst] += h[src].  One wave per edge; 32 lanes x float2
// covers the 64-dim feature row (coalesced 256B gather, f32 atomics scatter).
// ---------------------------------------------------------------------------
__global__ void edge_scatter_kernel(const float* __restrict__ hin, int lda,
                                    const int* __restrict__ src,
                                    const int* __restrict__ dst,
                                    float* __restrict__ agg, int n_edges) {
    int e = blockIdx.x * (blockDim.x >> 5) + (threadIdx.x >> 5);
    if (e >= n_edges) return;
    int lane = threadIdx.x & 31;
    int s = src[e];
    int d = dst[e];
    const float* hr = hin + (size_t)s * lda + lane * 2;
    float* ar = agg + (size_t)d * NODE_DIM + lane * 2;
    v2f v = *(const v2f*)hr;
    atomicAdd(ar + 0, v.x);
    atomicAdd(ar + 1, v.y);
}

// ---------------------------------------------------------------------------
// GIN layer GEMM: hout = relu(((1+eps)*hin + agg) @ W + b), W staged in LDS.
// Epilogue also accumulates the per-graph readout sum g[gid] += hout (L2 atomics).
// One wave computes a 16x64 tile via 16 K-steps x 4 N-tiles of WMMA f32 16x16x4.
// ---------------------------------------------------------------------------
__global__ void gin_gemm_kernel(const float* __restrict__ hin, int lda,
                                const float* __restrict__ agg,
                                const float* __restrict__ W,     // 64x64
                                const float* __restrict__ bias,  // 64
                                const float* __restrict__ eps, int layer,
                                float* __restrict__ hout, int ldc,
                                const int* __restrict__ gids,
                                float* __restrict__ g,
                                int n_nodes) {
    __shared__ float Wl[NODE_DIM * NODE_DIM];
    for (int i = threadIdx.x; i < NODE_DIM * NODE_DIM; i += blockDim.x)
        Wl[i] = W[i];
    __syncthreads();

    const float e1 = 1.0f + eps[layer];
    const int wave = threadIdx.x >> 5;
    const int lane = threadIdx.x & 31;
    const int rt = blockIdx.x * (blockDim.x >> 5) + wave;
    const int row0 = rt * 16;
    if (row0 >= n_nodes) return;

    const int m  = lane & 15;   // A-row / B-col within tile
    const int kh = lane >> 4;   // K-half select

    v8f acc0 = {}, acc1 = {}, acc2 = {}, acc3 = {};
    const float* arow = hin + (size_t)(row0 + m) * lda;
    const float* grow = agg + (size_t)(row0 + m) * NODE_DIM;

#pragma unroll
    for (int k0 = 0; k0 < NODE_DIM; k0 += 4) {
        const int ka = k0 + 2 * kh;
        v2f a;
        a.x = e1 * arow[ka]     + grow[ka];
        a.y = e1 * arow[ka + 1] + grow[ka + 1];
        const float* w0 = &Wl[ka * NODE_DIM + m];
        const float* w1 = &Wl[(ka + 1) * NODE_DIM + m];
        v2f b0, b1, b2, b3;
        b0.x = w0[0];  b0.y = w1[0];
        b1.x = w0[16]; b1.y = w1[16];
        b2.x = w0[32]; b2.y = w1[32];
        b3.x = w0[48]; b3.y = w1[48];
        acc0 = wmma_f32_k4(a, b0, acc0);
        acc1 = wmma_f32_k4(a, b1, acc1);
        acc2 = wmma_f32_k4(a, b2, acc2);
        acc3 = wmma_f32_k4(a, b3, acc3);
    }

    const float bv0 = bias[m];
    const float bv1 = bias[m + 16];
    const float bv2 = bias[m + 32];
    const float bv3 = bias[m + 48];

#pragma unroll
    for (int r = 0; r < 8; ++r) {
        const int row = row0 + r + 8 * kh;
        const int gid = gids[row];
        float* gr = g + (size_t)gid * (NODE_DIM * NUM_LAYERS) + layer * NODE_DIM;
        float* orow = hout + (size_t)row * ldc;

        float v0 = fmaxf(acc0[r] + bv0, 0.0f);
        float v1 = fmaxf(acc1[r] + bv1, 0.0f);
        float v2 = fmaxf(acc2[r] + bv2, 0.0f);
        float v3 = fmaxf(acc3[r] + bv3, 0.0f);
        orow[m]      = v0;
        orow[m + 16] = v1;
        orow[m + 32] = v2;
        orow[m + 48] = v3;
        atomicAdd(gr + m,      v0);
        atomicAdd(gr + m + 16, v1);
        atomicAdd(gr + m + 32, v2);
        atomicAdd(gr + m + 48, v3);
    }
}

// ---------------------------------------------------------------------------
// Generic small WMMA f32 GEMM: C = [relu](A[M,K] @ B[K,N] + bias).
// One wave per 16x16 output tile; waves flattened over (row-tile, col-tile).
// ---------------------------------------------------------------------------
__global__ void wmma_mlp_kernel(const float* __restrict__ A, int lda,
                                const float* __restrict__ B, int ldb,
                                const float* __restrict__ bias,
                                float* __restrict__ C, int ldc,
                                int M, int N, int K, int do_relu) {
    const int gwave = (blockIdx.x * blockDim.x + threadIdx.x) >> 5;
    const int tilesN = N >> 4;
    const int rt = gwave / tilesN;
    const int ct = gwave % tilesN;
    if (rt * 16 >= M) return;

    const int lane = threadIdx.x & 31;
    const int m  = lane & 15;
    const int kh = lane >> 4;

    v8f acc = {};
    const float* arow = A + (size_t)(rt * 16 + m) * lda;
    const float* bcol = B + ct * 16 + m;
    for (int k0 = 0; k0 < K; k0 += 4) {
        const int ka = k0 + 2 * kh;
        v2f a, b;
        a.x = arow[ka];
        a.y = arow[ka + 1];
        b.x = bcol[(size_t)ka * ldb];
        b.y = bcol[(size_t)(ka + 1) * ldb];
        acc = wmma_f32_k4(a, b, acc);
    }

    const int n = ct * 16 + m;
    const float bv = bias ? bias[n] : 0.0f;
#pragma unroll
    for (int r = 0; r < 8; ++r) {
        const int row = rt * 16 + r + 8 * kh;
        float v = acc[r] + bv;
        if (do_relu) v = fmaxf(v, 0.0f);
        C[(size_t)row * ldc + n] = v;
    }
}

// ---------------------------------------------------------------------------
// Launch: per layer { memset agg; edge scatter; WMMA GEMM (+readout atomics) }
// then two WMMA MLP kernels. All on `stream`, graph-capture safe.
// ---------------------------------------------------------------------------
extern "C" void kernel_launch(void* const* d_in, const int* in_sizes, int n_in,
                              void* d_out, int out_size, void* d_ws, size_t ws_size,
                              hipStream_t stream) {
    const float* x     = (const float*)d_in[0];
    const float* gin_W = (const float*)d_in[1];   // [3,64,64]
    const float* gin_b = (const float*)d_in[2];   // [3,64]
    const float* eps   = (const float*)d_in[3];   // [3]
    const float* r_W1  = (const float*)d_in[4];   // [192,128]
    const float* r_b1  = (const float*)d_in[5];   // [128]
    const float* r_W2  = (const float*)d_in[6];   // [128,32]
    const float* r_b2  = (const float*)d_in[7];   // [32]
    const int*   src   = (const int*)d_in[8];
    const int*   dst   = (const int*)d_in[9];
    const int*   gids  = (const int*)d_in[10];
    float* out = (float*)d_out;

    // Workspace layout
    const int LD_H = NODE_DIM * NUM_LAYERS;       // 192
    float* H      = (float*)d_ws;                              // N * 192
    float* agg    = H + (size_t)N_NODES * LD_H;                // N * 64
    float* g      = agg + (size_t)N_NODES * NODE_DIM;          // 256 * 192
    float* hidden = g + (size_t)NUM_GRAPHS * LD_H;             // 256 * 128

    hipMemsetAsync(g, 0, (size_t)NUM_GRAPHS * LD_H * sizeof(float), stream);

    const int scatter_blocks = (N_EDGES * 32 + 255) / 256;     // 8 edges/block
    const int row_tiles = N_NODES / 16;                        // 6250 (exact)
    const int gemm_blocks = (row_tiles + 7) / 8;               // 8 waves/block

    const float* hin = x;
    int lda = NODE_DIM;
    for (int l = 0; l < NUM_LAYERS; ++l) {
        hipMemsetAsync(agg, 0, (size_t)N_NODES * NODE_DIM * sizeof(float), stream);
        edge_scatter_kernel<<<scatter_blocks, 256, 0, stream>>>(
            hin, lda, src, dst, agg, N_EDGES);
        gin_gemm_kernel<<<gemm_blocks, 256, 0, stream>>>(
            hin, lda, agg,
            gin_W + (size_t)l * NODE_DIM * NODE_DIM,
            gin_b + (size_t)l * NODE_DIM,
            eps, l,
            H + (size_t)l * NODE_DIM, LD_H,
            gids, g, N_NODES);
        hin = H + (size_t)l * NODE_DIM;
        lda = LD_H;
    }

    // hidden = relu(g @ r_W1 + r_b1) : M=256, K=192, N=128 -> 16x8=128 waves
    wmma_mlp_kernel<<<16, 256, 0, stream>>>(
        g, LD_H, r_W1, R_HID, r_b1, hidden, R_HID,
        NUM_GRAPHS, R_HID, LD_H, 1);

    // out = hidden @ r_W2 + r_b2 : M=256, K=128, N=32 -> 16x2=32 waves
    wmma_mlp_kernel<<<4, 256, 0, stream>>>(
        hidden, R_HID, r_W2, R_OUT, r_b2, out, R_OUT,
        NUM_GRAPHS, R_OUT, R_HID, 0);
}